// Encoder_10926396801887
// MI455X (gfx1250) — compile-verified
//
#include <hip/hip_runtime.h>

#define HID 32
#define FK  35
#define OMEGA_ 100.0f
#define INV_PI 0.31830988618379067f

typedef float v8f __attribute__((ext_vector_type(8)));
typedef float v2f __attribute__((ext_vector_type(2)));

// ---------------- helpers ----------------
__device__ __forceinline__ int fkey(float f) {
    int i = __float_as_int(f);
    return (i >= 0) ? i : (i ^ 0x7fffffff);
}
__device__ __forceinline__ float funkey(int i) {
    return __int_as_float((i >= 0) ? i : (i ^ 0x7fffffff));
}
__device__ __forceinline__ float sel8(float p0, float p1, float p2, float p3,
                                      float p4, float p5, float p6, float p7, int r) {
    float a = (r & 1) ? p1 : p0;
    float b = (r & 1) ? p3 : p2;
    float c = (r & 1) ? p5 : p4;
    float d = (r & 1) ? p7 : p6;
    float e = (r & 2) ? b : a;
    float f = (r & 2) ? d : c;
    return (r & 4) ? f : e;
}

// ---------------- stats: x min / 1/range (cols 0..3), pos min / 1/range (cols 4..6) ----
__global__ void stats_kernel(const float* __restrict__ x, const float* __restrict__ pos,
                             int Nx, int Np, float* __restrict__ stats) {
    __shared__ float red[256];
    int tid = threadIdx.x;
    float mn[7], mx[7];
#pragma unroll
    for (int c = 0; c < 7; ++c) { mn[c] = 3.4e38f; mx[c] = -3.4e38f; }
    for (int i = tid; i < Nx; i += 256) {
#pragma unroll
        for (int f = 0; f < 4; ++f) {
            float v = x[i * 4 + f];
            mn[f] = fminf(mn[f], v); mx[f] = fmaxf(mx[f], v);
        }
    }
    for (int i = tid; i < Np; i += 256) {
#pragma unroll
        for (int d = 0; d < 3; ++d) {
            float v = pos[i * 3 + d];
            mn[4 + d] = fminf(mn[4 + d], v); mx[4 + d] = fmaxf(mx[4 + d], v);
        }
    }
    for (int c = 0; c < 7; ++c) {
        red[tid] = mn[c]; __syncthreads();
        for (int s = 128; s > 0; s >>= 1) { if (tid < s) red[tid] = fminf(red[tid], red[tid + s]); __syncthreads(); }
        float tmn = red[0]; __syncthreads();
        red[tid] = mx[c]; __syncthreads();
        for (int s = 128; s > 0; s >>= 1) { if (tid < s) red[tid] = fmaxf(red[tid], red[tid + s]); __syncthreads(); }
        float tmx = red[0]; __syncthreads();
        if (tid == 0) {
            if (c < 4) { stats[c] = tmn; stats[4 + c] = 1.0f / (tmx - tmn); }
            else       { stats[8 + (c - 4)] = tmn; stats[11 + (c - 4)] = 1.0f / (tmx - tmn); }
        }
        __syncthreads();
    }
}

// ---------------- lin0: h = sin(100*(xn @ W + b)) ----------------
__global__ void lin0_kernel(const float* __restrict__ x, const float* __restrict__ W,
                            const float* __restrict__ b, const float* __restrict__ stats,
                            float* __restrict__ out, int N) {
    int t = blockIdx.x * blockDim.x + threadIdx.x;
    if (t >= N * HID) return;
    int i = t >> 5, g = t & 31;
    float a = b[g];
#pragma unroll
    for (int f = 0; f < 4; ++f) {
        float xn = (x[i * 4 + f] - stats[f]) * stats[4 + f];
        a += xn * W[f * HID + g];
    }
    out[t] = __sinf(OMEGA_ * a);
}

// ---------------- conv: WMMA fp32, one wave per (edge|self-loop) ----------------
#define SLAB 1160  // H 32x33 (1056) + xj 40 + base 32 + w0d 32
__global__ __launch_bounds__(128, 1) void conv_kernel(
    const float* __restrict__ hin, const float* __restrict__ posr,
    const int* __restrict__ esrc, const int* __restrict__ edst,
    int E, int N,
    const float* __restrict__ W0, const float* __restrict__ b0,
    const float* __restrict__ W1, const float* __restrict__ b1,
    const float* __restrict__ W2, const float* __restrict__ b2,
    const float* __restrict__ Wl, const float* __restrict__ bl,
    const float* __restrict__ stats, float* __restrict__ acc) {
    __shared__ float lds[4 * SLAB];
    const int tid  = threadIdx.x;
    const int lane = tid & 31;
    const int wv   = tid >> 5;
    const int l15  = lane & 15;
    const int hi2  = (lane >> 4) << 1;   // K offset pattern (0 or 2)
    float* Hs   = lds + wv * SLAB;       // 32 x 33 (padded)
    float* xjs  = Hs + 1056;             // 35 (padded to 40)
    float* bas  = Hs + 1096;             // 32
    float* w0ds = Hs + 1128;             // 32

    // per-lane resident weights (shared across all edges this wave processes)
    const float w00 = W0[lane], w01 = W0[32 + lane], w02 = W0[64 + lane];
    const float w0d = W0[96 + lane], b0c = b0[lane];
    w0ds[lane] = w0d;
    float wlr[FK];
#pragma unroll
    for (int f = 0; f < FK; ++f) wlr[f] = Wl[lane * FK + f];
    const float bl_a = bl[lane];
    const float bl_b = (lane < 3) ? bl[32 + lane] : 0.0f;

    float w1B[2][8][2], w2B[2][8][2];   // B fragments (4x16 f32 tiles), resident
#pragma unroll
    for (int nt = 0; nt < 2; ++nt)
#pragma unroll
        for (int kt = 0; kt < 8; ++kt)
#pragma unroll
            for (int r = 0; r < 2; ++r) {
                int k = kt * 4 + hi2 + r, n = nt * 16 + l15;
                w1B[nt][kt][r] = W1[k * HID + n];
                w2B[nt][kt][r] = W2[k * HID + n];
            }
    const float b1v[2] = { b1[l15], b1[16 + l15] };
    const float b2v[2] = { b2[l15], b2[16 + l15] };
    const float pi0 = stats[11], pi1 = stats[12], pi2 = stats[13];

    const int EN = E + N;
    const int gw = blockIdx.x * 4 + wv;
    const int nw = gridDim.x * 4;
    for (int e = gw; e < EN; e += nw) {
        int s, d;
        float rho = 0.f, th = 0.f, ph = 0.f;
        if (e < E) {
            s = esrc[e]; d = edst[e];
            float rx = (posr[3 * d + 0] - posr[3 * s + 0]) * pi0;
            float ry = (posr[3 * d + 1] - posr[3 * s + 1]) * pi1;
            float rz = (posr[3 * d + 2] - posr[3 * s + 2]) * pi2;
            rho = sqrtf(rx * rx + ry * ry + rz * rz);
            if (rho > 0.f) {
                th = atan2f(ry, rx) * INV_PI;
                float zz = fminf(1.f, fmaxf(-1.f, rz / rho));
                ph = asinf(zz) * INV_PI;
            }
        } else { s = d = e - E; }

        // xj (35 feats) and base[h] into per-wave LDS slab
        xjs[lane] = hin[s * HID + lane];
        if (lane < 3) xjs[32 + lane] = (posr[3 * s + lane] - stats[8 + lane]) * stats[11 + lane];
        bas[lane] = rho * w00 + th * w01 + ph * w02 + b0c;
        asm volatile("s_wait_dscnt 0" ::: "memory");

        // v[h] = Wl[h,:]·xj  (lane h), s = bl·xj (wave-reduced)
        float v = 0.f;
#pragma unroll
        for (int f = 0; f < FK; ++f) v += wlr[f] * xjs[f];
        float sp = bl_a * xjs[lane] + ((lane < 3) ? bl_b * xjs[32 + lane] : 0.f);
        sp += __shfl_xor(sp, 16); sp += __shfl_xor(sp, 8); sp += __shfl_xor(sp, 4);
        sp += __shfl_xor(sp, 2);  sp += __shfl_xor(sp, 1);
        float v0 = __shfl(v, l15), v1 = __shfl(v, 16 + l15);

        // H0 A-fragments: H0[c,h] = sin(100*(base[h] + c*W0d[h]))
        float af[2][8][2];
#pragma unroll
        for (int kt = 0; kt < 8; ++kt)
#pragma unroll
            for (int r = 0; r < 2; ++r) {
                int h = kt * 4 + hi2 + r;
                float bb = bas[h], ww = w0ds[h];
                af[0][kt][r] = __sinf(OMEGA_ * (bb + (float)l15 * ww));
                af[1][kt][r] = __sinf(OMEGA_ * (bb + (float)(16 + l15) * ww));
            }

        // stage 1: H1 = sin(100*(H0 @ W1 + b1)) -> LDS (C-layout to A-layout bounce)
#pragma unroll
        for (int mt = 0; mt < 2; ++mt)
#pragma unroll
            for (int nt = 0; nt < 2; ++nt) {
                v8f cc = {0.f, 0.f, 0.f, 0.f, 0.f, 0.f, 0.f, 0.f};
#pragma unroll
                for (int kt = 0; kt < 8; ++kt) {
                    v2f av = { af[mt][kt][0], af[mt][kt][1] };
                    v2f bv = { w1B[nt][kt][0], w1B[nt][kt][1] };
                    cc = __builtin_amdgcn_wmma_f32_16x16x4_f32(false, av, false, bv,
                                                               (short)0, cc, false, false);
                }
#pragma unroll
                for (int j = 0; j < 8; ++j) {
                    int row = mt * 16 + j + (hi2 << 2);   // channel
                    Hs[row * 33 + nt * 16 + l15] = __sinf(OMEGA_ * (cc[j] + b1v[nt]));
                }
            }
        asm volatile("s_wait_dscnt 0" ::: "memory");
#pragma unroll
        for (int mt = 0; mt < 2; ++mt)
#pragma unroll
            for (int kt = 0; kt < 8; ++kt)
#pragma unroll
                for (int r = 0; r < 2; ++r)
                    af[mt][kt][r] = Hs[(mt * 16 + l15) * 33 + kt * 4 + hi2 + r];

        // stage 2: H2 = sin(100*(H1 @ W2 + b2)); fuse dot with v in C-layout
        float part[2][8];
#pragma unroll
        for (int mt = 0; mt < 2; ++mt) {
#pragma unroll
            for (int j = 0; j < 8; ++j) part[mt][j] = 0.f;
#pragma unroll
            for (int nt = 0; nt < 2; ++nt) {
                v8f cc = {0.f, 0.f, 0.f, 0.f, 0.f, 0.f, 0.f, 0.f};
#pragma unroll
                for (int kt = 0; kt < 8; ++kt) {
                    v2f av = { af[mt][kt][0], af[mt][kt][1] };
                    v2f bv = { w2B[nt][kt][0], w2B[nt][kt][1] };
                    cc = __builtin_amdgcn_wmma_f32_16x16x4_f32(false, av, false, bv,
                                                               (short)0, cc, false, false);
                }
                float vn = nt ? v1 : v0;
#pragma unroll
                for (int j = 0; j < 8; ++j)
                    part[mt][j] += __sinf(OMEGA_ * (cc[j] + b2v[nt])) * vn;
            }
#pragma unroll
            for (int j = 0; j < 8; ++j) {
                part[mt][j] += __shfl_xor(part[mt][j], 1);
                part[mt][j] += __shfl_xor(part[mt][j], 2);
                part[mt][j] += __shfl_xor(part[mt][j], 4);
                part[mt][j] += __shfl_xor(part[mt][j], 8);
            }
        }
        // lane `lane` emits channel `lane`
        int rr = lane & 7;
        float a0 = sel8(part[0][0], part[0][1], part[0][2], part[0][3],
                        part[0][4], part[0][5], part[0][6], part[0][7], rr);
        float a1 = sel8(part[1][0], part[1][1], part[1][2], part[1][3],
                        part[1][4], part[1][5], part[1][6], part[1][7], rr);
        float o0 = __shfl_xor(a0, 16), o1 = __shfl_xor(a1, 16);
        float val;
        if (lane < 16) val = (lane & 8) ? o0 : a0;
        else           val = (lane & 8) ? a1 : o1;
        val += sp;
        atomicAdd(acc + d * HID + lane, val);
    }
}

// ---------------- sin(100*(acc + bias)) ----------------
__global__ void sinbias_kernel(const float* __restrict__ acc, const float* __restrict__ bias,
                               float* __restrict__ out, int n) {
    int t = blockIdx.x * blockDim.x + threadIdx.x;
    if (t < n) out[t] = __sinf(OMEGA_ * (acc[t] + bias[t & 31]));
}

// ---------------- max-pool via ordered int keys ----------------
__global__ void poolinit_kernel(const float* __restrict__ h, int* __restrict__ key, int n) {
    int t = blockIdx.x * blockDim.x + threadIdx.x;
    if (t < n) key[t] = fkey(h[t]);
}
__global__ void pooledge_kernel(const float* __restrict__ h, const int* __restrict__ esrc,
                                const int* __restrict__ edst, int* __restrict__ key, int E) {
    int t = blockIdx.x * blockDim.x + threadIdx.x;
    if (t >= E * HID) return;
    int e = t >> 5, c = t & 31;
    atomicMax(&key[edst[e] * HID + c], fkey(h[esrc[e] * HID + c]));
}
__global__ void poolgather_kernel(const int* __restrict__ key, const int* __restrict__ keep,
                                  float* __restrict__ out, int n) {
    int t = blockIdx.x * blockDim.x + threadIdx.x;
    if (t >= n) return;
    int j = t >> 5, c = t & 31;
    out[t] = funkey(key[keep[j] * HID + c]);
}

// ---------------- final: mean over nodes then lin1 ----------------
__global__ void final_kernel(const float* __restrict__ h, int N,
                             const float* __restrict__ W, const float* __restrict__ b,
                             float* __restrict__ out) {
    __shared__ float red[256];
    __shared__ float colm[32];
    int tid = threadIdx.x;
    int c = tid & 31;
    float s = 0.f;
    for (int i = tid >> 5; i < N; i += 8) s += h[i * HID + c];
    red[tid] = s; __syncthreads();
    if (tid < 32) {
        float m = 0.f;
#pragma unroll
        for (int w = 0; w < 8; ++w) m += red[w * 32 + tid];
        colm[tid] = m / (float)N;
    }
    __syncthreads();
    if (tid < 32) {
        float a = b[tid];
#pragma unroll
        for (int k = 0; k < 32; ++k) a += colm[k] * W[k * 32 + tid];
        out[tid] = a;
    }
}

// ---------------- launch ----------------
extern "C" void kernel_launch(void* const* d_in, const int* in_sizes, int n_in,
                              void* d_out, int out_size, void* d_ws, size_t ws_size,
                              hipStream_t stream) {
    (void)in_sizes; (void)n_in; (void)out_size; (void)ws_size;
    const float* x    = (const float*)d_in[0];
    const float* pos  = (const float*)d_in[1];
    const int*   ei0  = (const int*)d_in[2];
    const int*   ei1  = (const int*)d_in[3];
    const int*   ei2  = (const int*)d_in[4];
    const int*   ei3  = (const int*)d_in[5];
    const float* pp1  = (const float*)d_in[6];
    const float* pp2  = (const float*)d_in[7];
    const float* pp3  = (const float*)d_in[8];
    const int*   ki0  = (const int*)d_in[9];
    const int*   ki1  = (const int*)d_in[10];
    const int*   ki2  = (const int*)d_in[11];
    const float* lin0W = (const float*)d_in[12];
    const float* lin0b = (const float*)d_in[13];
    const float* kW0 = (const float*)d_in[14];
    const float* kb0 = (const float*)d_in[15];
    const float* kW1 = (const float*)d_in[16];
    const float* kb1 = (const float*)d_in[17];
    const float* kW2 = (const float*)d_in[18];
    const float* kb2 = (const float*)d_in[19];
    const float* kWl = (const float*)d_in[20];
    const float* kbl = (const float*)d_in[21];
    const float* cbias = (const float*)d_in[22];
    const float* lin1W = (const float*)d_in[23];
    const float* lin1b = (const float*)d_in[24];

    const int NS[4] = {10000, 5000, 2500, 1250};
    const int ES[4] = {40000, 20000, 10000, 5000};
    const float* posL[4] = {pos, pp1, pp2, pp3};
    const int*   eiL[4]  = {ei0, ei1, ei2, ei3};
    const int*   keepL[3] = {ki0, ki1, ki2};

    float* ws    = (float*)d_ws;
    float* stats = ws;               // 16 floats
    float* A     = ws + 16;          // 320000 floats (node features, ping)
    float* B     = A + 320000;       // 320000 floats (conv accum / pool keys, pong)

    stats_kernel<<<1, 256, 0, stream>>>(x, pos, NS[0], NS[0], stats);
    {
        int n = NS[0] * HID;
        lin0_kernel<<<(n + 255) / 256, 256, 0, stream>>>(x, lin0W, lin0b, stats, A, NS[0]);
    }
    for (int l = 0; l < 4; ++l) {
        int N = NS[l], E = ES[l];
        hipMemsetAsync(B, 0, (size_t)N * HID * sizeof(float), stream);
        int EN = E + N;
        conv_kernel<<<(EN + 3) / 4, 128, 0, stream>>>(
            A, posL[l], eiL[l], eiL[l] + E, E, N,
            kW0 + l * 128, kb0 + l * 32, kW1 + l * 1024, kb1 + l * 32,
            kW2 + l * 1024, kb2 + l * 32, kWl + l * (HID * FK), kbl + l * FK,
            stats, B);
        int n = N * HID;
        sinbias_kernel<<<(n + 255) / 256, 256, 0, stream>>>(B, cbias + l * 32, A, n);
        if (l < 3) {
            poolinit_kernel<<<(n + 255) / 256, 256, 0, stream>>>(A, (int*)B, n);
            int ne = E * HID;
            pooledge_kernel<<<(ne + 255) / 256, 256, 0, stream>>>(A, eiL[l], eiL[l] + E, (int*)B, E);
            int nn = NS[l + 1] * HID;
            poolgather_kernel<<<(nn + 255) / 256, 256, 0, stream>>>((int*)B, keepL[l], A, nn);
        }
    }
    final_kernel<<<1, 256, 0, stream>>>(A, NS[3], lin1W, lin1b, (float*)d_out);
}